// Loss_functions_7748121002349
// MI455X (gfx1250) — compile-verified
//
#include <hip/hip_runtime.h>

typedef float v2f __attribute__((ext_vector_type(2)));
typedef float v8f __attribute__((ext_vector_type(8)));

#define D_MIN_F 0.001f
#define NBINS 256
#define NTILES 16           // 256 centers / 16 rows per WMMA tile
#define GPW 32              // point-groups (of 16) per wave
#define WPB 8               // waves per block (256 threads)
#define PTS_PER_BLOCK (WPB * GPW * 16)   // 4096 points per block

// ---------------- workspace init ----------------
__global__ void init_ws(double* S, double* dysum, unsigned int* len, unsigned int* rowmin) {
  int i = threadIdx.x;
  if (i < 4 * NBINS) rowmin[i] = 0x7F7FFFFFu;   // +FLT_MAX bits
  if (i == 0) {
    S[0] = 0.0; S[1] = 0.0;
    for (int k = 0; k < 4; ++k) { dysum[k] = 0.0; len[k] = 0u; }
  }
}

// ---------------- SILog partial sums: S1 = sum g, S2 = sum g^2 ----------------
__global__ void silog_kernel(const float* __restrict__ pr, const float* __restrict__ tg,
                             double* S, int N) {
  __shared__ float sh1[256];
  __shared__ float sh2[256];
  int tid = threadIdx.x;
  int i = blockIdx.x * 256 + tid;
  float g = 0.0f;
  if (i < N) {
    float p = pr[i], t = tg[i];
    if (p >= D_MIN_F && t >= D_MIN_F)
      g = logf(p + 1e-5f) - logf(t + 1e-5f);   // masked entries contribute exactly 0
  }
  sh1[tid] = g;
  sh2[tid] = g * g;
  __syncthreads();
  for (int s = 128; s > 0; s >>= 1) {
    if (tid < s) { sh1[tid] += sh1[tid + s]; sh2[tid] += sh2[tid + s]; }
    __syncthreads();
  }
  if (tid == 0) {
    atomicAdd(&S[0], (double)sh1[0]);
    atomicAdd(&S[1], (double)sh2[0]);
  }
}

// ---------------- 1-D Chamfer via f32 WMMA distance tiles ----------------
// d[m][n] = (c_m - p_n)^2 = [c^2, -2c, 1, 0] . [1, p, p^2, 0]^T  -> 16x16x4 f32 WMMA
__global__ __launch_bounds__(256) void chamfer_kernel(
    const float* __restrict__ pts,       // [B, P] point cloud (target or lidar)
    const float* __restrict__ centers,   // [B, NBINS]
    unsigned int* __restrict__ rowmin_g, // [B, NBINS] float bits, atomic-min
    double* __restrict__ dysum_g,        // [B] sum of per-valid-point min distances
    unsigned int* __restrict__ len_g,    // [B] valid point counts
    int P) {
  const int b = blockIdx.y;
  const int tid = threadIdx.x;
  const int lane = tid & 31;
  const int wave = tid >> 5;
  const int lane15 = lane & 15;
  const bool lo = lane < 16;

  __shared__ unsigned int s_rowmin[NBINS];
  for (int i = tid; i < NBINS; i += 256) s_rowmin[i] = 0x7F7FFFFFu;
  __syncthreads();

  // Build A tiles from centers. 32-bit A 16x4 layout: lanes 0-15 hold K=0 (v0) and
  // K=1 (v1) for row M=lane; lanes 16-31 hold K=2 (v0) and K=3 (v1).
  const float* cb = centers + b * NBINS;
  v2f A[NTILES];
#pragma unroll
  for (int t = 0; t < NTILES; ++t) {
    float c = lo ? cb[t * 16 + lane15] : 0.0f;
    A[t].x = lo ? c * c : 1.0f;        // A[m][0] = c^2 ; A[m][2] = 1
    A[t].y = lo ? -2.0f * c : 0.0f;    // A[m][1] = -2c ; A[m][3] = 0
  }

  // Row-min accumulators: D-layout keeps row M = t*16 + (lo ? r : r+8) fixed per
  // (vgpr r, lane half), so elementwise min across point groups is valid.
  v8f rowacc[NTILES];
#pragma unroll
  for (int t = 0; t < NTILES; ++t)
#pragma unroll
    for (int r = 0; r < 8; ++r) rowacc[t][r] = 3.0e38f;

  float dy_sum = 0.0f;
  float vcount = 0.0f;

  const float* pb = pts + (long long)b * P;
  const int gwave = blockIdx.x * WPB + wave;
  const int base0 = gwave * (GPW * 16);

  for (int gIdx = 0; gIdx < GPW; ++gIdx) {
    const int pbase = base0 + gIdx * 16;
    if (pbase >= P) break;                    // wave-uniform branch, EXEC stays full
    float p = pb[pbase + lane15];             // lanes l and l+16 load the same point
    bool valid = (p >= D_MIN_F);
    float pe = valid ? p : 1.0e5f;            // invalid -> distance ~1e10 (BIG)
    // B 4x16 layout mirrors A: lanes 0-15 rows K=0 (v0), K=1 (v1); lanes 16-31 K=2,K=3
    v2f Bm;
    Bm.x = lo ? 1.0f : pe * pe;               // B[0][n] = 1 ; B[2][n] = p^2
    Bm.y = lo ? pe : 0.0f;                    // B[1][n] = p ; B[3][n] = 0

    float cmin = 3.0e38f;
#pragma unroll
    for (int t = 0; t < NTILES; ++t) {
      v8f dacc = {0.f, 0.f, 0.f, 0.f, 0.f, 0.f, 0.f, 0.f};
      // (neg_a, A, neg_b, B, c_mod, C, reuse_a, reuse_b)
      dacc = __builtin_amdgcn_wmma_f32_16x16x4_f32(false, A[t], false, Bm,
                                                   (short)0, dacc, false, false);
#pragma unroll
      for (int r = 0; r < 8; ++r) {
        float d = dacc[r];
        rowacc[t][r] = fminf(rowacc[t][r], d);   // row (center) min accumulate
        cmin = fminf(cmin, d);                   // partial column (point) min
      }
    }
    // full column min = min over both row-halves
    float full = fminf(cmin, __shfl_xor(cmin, 16, 32));
    full = fmaxf(full, 0.0f);
    float take = (lo && valid) ? 1.0f : 0.0f;    // count each point once
    dy_sum += take * full;
    vcount += take;
  }

  // wave-reduce point-side sums (lanes >= 16 contribute zeros)
#pragma unroll
  for (int m = 16; m >= 1; m >>= 1) {
    dy_sum += __shfl_xor(dy_sum, m, 32);
    vcount += __shfl_xor(vcount, m, 32);
  }
  if (lane == 0) {
    atomicAdd(&dysum_g[b], (double)dy_sum);
    atomicAdd(&len_g[b], (unsigned int)(vcount + 0.5f));
  }

  // row-min: butterfly min within each 16-lane half (xor 1,2,4,8 stays in-half)
#pragma unroll
  for (int t = 0; t < NTILES; ++t) {
#pragma unroll
    for (int r = 0; r < 8; ++r) {
      float v = rowacc[t][r];
      v = fminf(v, __shfl_xor(v, 1, 32));
      v = fminf(v, __shfl_xor(v, 2, 32));
      v = fminf(v, __shfl_xor(v, 4, 32));
      v = fminf(v, __shfl_xor(v, 8, 32));
      v = fmaxf(v, 0.0f);                      // keep uint-ordering trick sound
      unsigned int bits = __float_as_uint(v);
      if (lane == 0)  atomicMin(&s_rowmin[t * 16 + r], bits);       // row M = t*16+r
      if (lane == 16) atomicMin(&s_rowmin[t * 16 + r + 8], bits);   // row M = t*16+r+8
    }
  }
  __syncthreads();
  for (int i = tid; i < NBINS; i += 256)
    atomicMin(&rowmin_g[b * NBINS + i], s_rowmin[i]);
}

// ---------------- final scalar combine ----------------
__global__ void final_kernel(const double* S, const double* dysum, const unsigned int* len,
                             const unsigned int* rowmin, float* out, int N) {
  if (threadIdx.x != 0 || blockIdx.x != 0) return;
  double s1 = S[0], s2 = S[1];
  double nn = (double)N;
  double mean = s1 / nn;
  double var = (s2 - s1 * s1 / nn) / (nn - 1.0);   // ddof=1 over all N entries
  double sil = sqrt(var + (1.0 - 0.85) * mean * mean);
  double bcl[2];
  for (int c2 = 0; c2 < 2; ++c2) {
    double acc = 0.0;
    for (int b = 0; b < 2; ++b) {
      int idx = c2 * 2 + b;
      double rx = 0.0;
      for (int m = 0; m < NBINS; ++m)
        rx += (double)__uint_as_float(rowmin[idx * NBINS + m]);
      rx /= (double)NBINS;                         // cham_x: mean over bins
      unsigned int L = len[idx];
      double ry = dysum[idx] / (double)(L > 0u ? L : 1u);  // cham_y: mean over valid pts
      acc += rx + ry;
    }
    bcl[c2] = acc * 0.5;                           // batch mean
  }
  out[0] = (float)(10.0 * sil + 0.1 * bcl[0] + 0.001 * bcl[1]);
}

extern "C" void kernel_launch(void* const* d_in, const int* in_sizes, int n_in,
                              void* d_out, int out_size, void* d_ws, size_t ws_size,
                              hipStream_t stream) {
  (void)n_in; (void)out_size; (void)ws_size;
  const float* predict = (const float*)d_in[0];
  const float* centers = (const float*)d_in[1];
  const float* target  = (const float*)d_in[2];
  const float* lidar   = (const float*)d_in[3];
  float* out = (float*)d_out;

  const int N = in_sizes[0];    // B*H*W = 614400
  const int B = 2;
  const int P = N / B;          // 307200

  char* ws = (char*)d_ws;
  double* S            = (double*)(ws);          // 2 doubles: S1, S2
  double* dysum        = (double*)(ws + 16);     // 4 doubles: [cloud*2+b]
  unsigned int* len    = (unsigned int*)(ws + 48);  // 4 uints
  unsigned int* rowmin = (unsigned int*)(ws + 64);  // 4*256 uints (float bits)

  init_ws<<<1, 1024, 0, stream>>>(S, dysum, len, rowmin);

  silog_kernel<<<(N + 255) / 256, 256, 0, stream>>>(predict, target, S, N);

  dim3 grid((P + PTS_PER_BLOCK - 1) / PTS_PER_BLOCK, B);
  chamfer_kernel<<<grid, 256, 0, stream>>>(target, centers, rowmin + 0 * 2 * NBINS,
                                           dysum + 0, len + 0, P);
  chamfer_kernel<<<grid, 256, 0, stream>>>(lidar, centers, rowmin + 1 * 2 * NBINS,
                                           dysum + 2, len + 2, P);

  final_kernel<<<1, 32, 0, stream>>>(S, dysum, len, rowmin, out, N);
}